// ModuleWithRouting_61031485276532
// MI455X (gfx1250) — compile-verified
//
#include <hip/hip_runtime.h>
#include <stdint.h>

// ---------------------------------------------------------------------------
// MoE expert-0 routing mask (keep row iff expert 0 is in top-2 of 8).
//
// Streaming problem: 128 MB in + 128 MB out => ~11 us floor @ 23.3 TB/s.
// gfx1250 strategy:
//   * per-lane async global->LDS staging (GLOBAL_LOAD_ASYNC_TO_LDS_B128),
//     double-buffered, synchronized with s_wait_asynccnt only (each lane
//     consumes only the LDS bytes it staged itself -> no barriers).
//   * NT load hints + non-temporal stores (stream > 192 MB L2, zero reuse).
//   * compile-time lane count + host-computed trip count: no device-side
//     integer division, minimal SALU prologue.
// ---------------------------------------------------------------------------

typedef float v4f __attribute__((ext_vector_type(4)));

#define BLOCK 256
#define GRID_BLOCKS 4096
#define E 8                                   // experts per row (32 B rows)
constexpr int TOTAL = BLOCK * GRID_BLOCKS;    // lanes in the grid (1M)
constexpr long long ROW_STRIDE_BYTES = (long long)TOTAL * E * sizeof(float);

// Stage one 32 B row as two async 16 B global->LDS copies. IOFFSET is added
// to BOTH the LDS and the global address (ISA 08 §4.4), so offset:16 covers
// the second half of the row on both sides with one address pair.
__device__ __forceinline__ void async_stage_row(uint32_t lds_off, uint64_t gaddr) {
    asm volatile(
        "global_load_async_to_lds_b128 %0, %1, off th:TH_LOAD_NT\n\t"
        "global_load_async_to_lds_b128 %0, %1, off offset:16 th:TH_LOAD_NT"
        :
        : "v"(lds_off), "v"(gaddr)
        : "memory");
}

__global__ __launch_bounds__(BLOCK) void moe_route_e0_kernel(
    const float* __restrict__ x, float* __restrict__ out, int rows, int niter) {
    // Per-thread staging slot: 8 floats, double buffered. 16 KB LDS / block.
    __shared__ __align__(16) float tile[2][BLOCK * E];

    const int tid = (int)threadIdx.x;
    const int g0  = (int)blockIdx.x * BLOCK + tid; // this lane's first row

    // LDS byte offsets of the two staging slots: low 32 bits of a generic
    // pointer to a __shared__ object are the LDS address.
    const uint32_t lds0 = (uint32_t)(uintptr_t)(&tile[0][tid * E]);
    const uint32_t lds1 = (uint32_t)(uintptr_t)(&tile[1][tid * E]);

    const int last_row = rows - 1;
    auto row_addr = [&](int r) {
        r = (r < rows) ? r : last_row; // clamp prefetch, never read OOB
        return (uint64_t)(uintptr_t)(x + (size_t)r * E);
    };

    // Prefetch iteration 0 into buffer 0.
    async_stage_row(lds0, row_addr(g0));

    for (int it = 0; it < niter; ++it) {
        const int      cur     = it & 1;
        const uint32_t lds_nxt = cur ? lds0 : lds1;

        if (it + 1 < niter) {
            // Kick off the next tile, then wait for the 2 older copies to
            // land: ASYNCcnt completes in order, so <=2 releases the current
            // buffer while the new pair stays in flight.
            async_stage_row(lds_nxt, row_addr(g0 + (it + 1) * TOTAL));
            asm volatile("s_wait_asynccnt 0x2" ::: "memory");
        } else {
            asm volatile("s_wait_asynccnt 0x0" ::: "memory");
        }

        // Consume this lane's staged row from LDS (2 x ds_load_b128).
        const v4f* src = (const v4f*)(&tile[cur][tid * E]);
        v4f a = src[0];
        v4f b = src[1];

        // Expert 0 survives top-2 (lower-index tie-break) iff at most one
        // other expert strictly beats it.
        const float x0 = a.x;
        int cnt = (a.y > x0) + (a.z > x0) + (a.w > x0) +
                  (b.x > x0) + (b.y > x0) + (b.z > x0) + (b.w > x0);
        const bool keep = (cnt <= 1);

        const v4f zero = {0.0f, 0.0f, 0.0f, 0.0f};
        v4f oa = keep ? a : zero;
        v4f ob = keep ? b : zero;

        const int row = g0 + it * TOTAL;
        if (row < rows) {
            v4f* o = (v4f*)(out + (size_t)row * E);
            __builtin_nontemporal_store(oa, o);
            __builtin_nontemporal_store(ob, o + 1);
        }
    }
}

extern "C" void kernel_launch(void* const* d_in, const int* in_sizes, int n_in,
                              void* d_out, int out_size, void* d_ws, size_t ws_size,
                              hipStream_t stream) {
    (void)n_in; (void)out_size; (void)d_ws; (void)ws_size;
    const float* x   = (const float*)d_in[0];
    float*       out = (float*)d_out;
    const int rows  = in_sizes[0] / E;               // [rows, 8] float32
    if (rows <= 0) return;
    const int niter = (rows + TOTAL - 1) / TOTAL;    // host-side: no device div

    // 4096 blocks x 256 threads = 1M lanes -> 4 rows/lane at SEQ_LEN = 4M,
    // uniform trip count for exact async-counter bookkeeping in every wave.
    moe_route_e0_kernel<<<dim3(GRID_BLOCKS), dim3(BLOCK), 0, stream>>>(x, out, rows, niter);
}